// ProgressiveThoughtEncoder_10883447128196
// MI455X (gfx1250) — compile-verified
//
#include <hip/hip_runtime.h>
#include <hip/hip_bf16.h>
#include <math.h>

typedef __attribute__((ext_vector_type(2))) float v2f;
typedef __attribute__((ext_vector_type(8))) float v8f;

#define Bsz 2
#define Lsz 32
#define Esz 256
#define Hsz 4096
#define Gsz 16
#define Rsz 64
#define CHUNKS 16

// ---------------------------------------------------------------------------
// Kernel 1: wk_q[h] = sum_r Wk[r,h] * layer_query[r]
// ---------------------------------------------------------------------------
__global__ void wkq_kernel(const float* __restrict__ Wk,
                           const float* __restrict__ lq,
                           float* __restrict__ wkq) {
    int h = blockIdx.x * blockDim.x + threadIdx.x;
    float acc = 0.0f;
#pragma unroll 8
    for (int r = 0; r < Rsz; ++r) acc += Wk[r * Hsz + h] * lq[r];
    wkq[h] = acc;
}

// ---------------------------------------------------------------------------
// Kernel 2: partial[(b*L+l)*CHUNKS + c] = sum over chunk of K[b,l,e,h]*wkq[h]
// grid = (CHUNKS, B*L), block = 256. Each slice (b,l) is E*H = 1M floats.
// ---------------------------------------------------------------------------
__global__ void score_partial_kernel(const float* __restrict__ K,
                                     const float* __restrict__ wkq,
                                     float* __restrict__ partial) {
    const int c  = blockIdx.x;
    const int sl = blockIdx.y;            // b*L + l
    const int t  = threadIdx.x;
    __shared__ float wq[Hsz];
    __shared__ float red[256];
    for (int i = t; i < Hsz; i += 256) wq[i] = wkq[i];
    __syncthreads();

    const float4* K4 = (const float4*)(K + (size_t)sl * (Esz * Hsz));
    const int base4 = c * (65536 / 4);    // chunk of 65536 floats
    float acc = 0.0f;
#pragma unroll 4
    for (int i = 0; i < 64; ++i) {
        int f4 = base4 + i * 256 + t;     // float4 index within slice
        float4 kv = K4[f4];
        int h = (f4 << 2) & (Hsz - 1);
        acc += kv.x * wq[h] + kv.y * wq[h + 1] + kv.z * wq[h + 2] + kv.w * wq[h + 3];
    }
    red[t] = acc; __syncthreads();
    for (int o = 128; o > 0; o >>= 1) {
        if (t < o) red[t] += red[t + o];
        __syncthreads();
    }
    if (t == 0) partial[sl * CHUNKS + c] = red[0];
}

// ---------------------------------------------------------------------------
// Kernel 3: per-batch softmax over L. 1 block, 64 threads (t = b*32 + l)
// score = (1/E) * sum * inv_sqrt_r  (LAYER_TEMP = 1)
// ---------------------------------------------------------------------------
__global__ void layer_softmax_kernel(const float* __restrict__ partial,
                                     float* __restrict__ w) {
    const int t = threadIdx.x;
    float s = 0.0f;
#pragma unroll
    for (int c = 0; c < CHUNKS; ++c) s += partial[t * CHUNKS + c];
    s *= (0.125f / (float)Esz);           // inv_sqrt_r = 1/8, mean 1/E
    __shared__ float sc[Bsz * Lsz];
    sc[t] = s; __syncthreads();
    const int b = t >> 5;
    float mx = -1e30f;
    for (int l = 0; l < Lsz; ++l) mx = fmaxf(mx, sc[b * Lsz + l]);
    float e = __expf(s - mx);
    __syncthreads(); sc[t] = e; __syncthreads();
    float sum = 0.0f;
    for (int l = 0; l < Lsz; ++l) sum += sc[b * Lsz + l];
    w[t] = e / sum;
}

// ---------------------------------------------------------------------------
// Kernel 4: ek[b,e,h] = sum_l w[b,l]*K[b,l,e,h]; same for ev. float4 streaming.
// grid = 2048 blocks * 256 thr; each thread one float4 position of [B,E*H/4].
// ---------------------------------------------------------------------------
__global__ void weighted_sum_kernel(const float* __restrict__ K,
                                    const float* __restrict__ V,
                                    const float* __restrict__ w,
                                    float* __restrict__ ek,
                                    float* __restrict__ ev) {
    const int per_b = (Esz * Hsz) / 4;    // 262144 float4s per batch
    const int p = blockIdx.x * 256 + threadIdx.x;
    const int b = p / per_b;              // uniform per block (1024 blocks / b)
    const int rem = p - b * per_b;
    __shared__ float wl[Lsz];
    if (threadIdx.x < Lsz) wl[threadIdx.x] = w[b * Lsz + threadIdx.x];
    __syncthreads();
    const float4* K4 = (const float4*)K;
    const float4* V4 = (const float4*)V;
    const size_t base = (size_t)b * Lsz * per_b + rem;
    float4 ak = {0, 0, 0, 0}, av = {0, 0, 0, 0};
#pragma unroll 4
    for (int l = 0; l < Lsz; ++l) {
        float wv = wl[l];
        float4 kk = K4[base + (size_t)l * per_b];
        float4 vv = V4[base + (size_t)l * per_b];
        ak.x += wv * kk.x; ak.y += wv * kk.y; ak.z += wv * kk.z; ak.w += wv * kk.w;
        av.x += wv * vv.x; av.y += wv * vv.y; av.z += wv * vv.z; av.w += wv * vv.w;
    }
    ((float4*)ek)[p] = ak;
    ((float4*)ev)[p] = av;
}

// ---------------------------------------------------------------------------
// Kernel 5: Out[rows x R] = A[rows x H] * W^T  via V_WMMA_F32_16X16X4_F32.
// grid.x = rows/16; block = 128 (4 waves, one 16-wide r-tile each).
// ISA fragment layouts (wave32):
//   A 16x4 f32 : lane l -> {m=l%16, k=2*(l/16)+j}, j=0,1  (2 VGPRs)
//   B 4x16 f32 : lane l -> {k=2*(l/16)+j, n=l%16}         (2 VGPRs)
//   C/D 16x16  : lane l, vgpr j -> row = j + 8*(l/16), col = l%16
// ---------------------------------------------------------------------------
__global__ void proj_wmma_kernel(const float* __restrict__ A,
                                 const float* __restrict__ W,
                                 float* __restrict__ out) {
    const int tile = blockIdx.x;
    const int lane = threadIdx.x & 31;
    const int wave = threadIdx.x >> 5;
    const int r0 = wave * 16;
    const int mn = lane & 15;
    const int koff = (lane >> 4) << 1;
    const float* arow = A + (size_t)(tile * 16 + mn) * Hsz + koff;
    const float* brow = W + (size_t)(r0 + mn) * Hsz + koff;  // B[k,n] = W[n,k]
    v8f c = {0.f, 0.f, 0.f, 0.f, 0.f, 0.f, 0.f, 0.f};
#pragma unroll 4
    for (int h = 0; h < Hsz; h += 4) {
        v2f a = *(const v2f*)(arow + h);
        v2f b = *(const v2f*)(brow + h);
        c = __builtin_amdgcn_wmma_f32_16x16x4_f32(
                /*neg_a=*/false, a, /*neg_b=*/false, b,
                /*c_mod=*/(short)0, c, /*reuse_a=*/false, /*reuse_b=*/false);
    }
    const int mbase = tile * 16 + ((lane >> 4) << 3);
#pragma unroll
    for (int j = 0; j < 8; ++j)
        out[(size_t)(mbase + j) * Rsz + r0 + mn] = c[j];
}

// ---------------------------------------------------------------------------
// Kernel 6: per (b,g) attention: softmax_E(q·k * 1/8) @ v -> ctx[b,g,r]
// grid = B*G blocks, 256 threads (thread = e)
// ---------------------------------------------------------------------------
__global__ void attn_kernel(const float* __restrict__ q,
                            const float* __restrict__ kmat,
                            const float* __restrict__ vmat,
                            float* __restrict__ ctx) {
    const int bg = blockIdx.x;
    const int b = bg >> 4, g = bg & 15;
    const int t = threadIdx.x;
    __shared__ float qs[Rsz];
    __shared__ float attn[Esz];
    __shared__ float red[256];
    if (t < Rsz) qs[t] = q[g * Rsz + t];
    __syncthreads();
    const float* krow = kmat + (size_t)(b * Esz + t) * Rsz;
    float s = 0.0f;
#pragma unroll 8
    for (int r = 0; r < Rsz; ++r) s += qs[r] * krow[r];
    s *= 0.125f;
    red[t] = s; __syncthreads();
    for (int o = 128; o > 0; o >>= 1) {
        if (t < o) red[t] = fmaxf(red[t], red[t + o]);
        __syncthreads();
    }
    float mx = red[0]; __syncthreads();
    float e = __expf(s - mx);
    red[t] = e; __syncthreads();
    for (int o = 128; o > 0; o >>= 1) {
        if (t < o) red[t] += red[t + o];
        __syncthreads();
    }
    attn[t] = e / red[0];
    __syncthreads();
    if (t < Rsz) {
        float acc = 0.0f;
        for (int e2 = 0; e2 < Esz; ++e2)
            acc += attn[e2] * vmat[(size_t)(b * Esz + e2) * Rsz + t];
        ctx[bg * Rsz + t] = acc;
    }
}

// ---------------------------------------------------------------------------
// Kernel 7: s_new = mean_g ctx; gated update + norm clamp. grid=B, block=R.
// ---------------------------------------------------------------------------
__global__ void state_update_kernel(const float* __restrict__ ctx,
                                    const float* __restrict__ state,
                                    const int* __restrict__ ept,
                                    float* __restrict__ out) {
    const int b = blockIdx.x, r = threadIdx.x;
    float snew = 0.0f;
#pragma unroll
    for (int g = 0; g < Gsz; ++g) snew += ctx[(b * Gsz + g) * Rsz + r];
    snew *= (1.0f / (float)Gsz);
    float prev = state[b * Rsz + r];
    __shared__ float rp[Rsz], rs[Rsz], rd[Rsz];
    rp[r] = prev * prev; rs[r] = snew * snew; rd[r] = prev * snew;
    __syncthreads();
    for (int o = 32; o > 0; o >>= 1) {
        if (r < o) { rp[r] += rp[r + o]; rs[r] += rs[r + o]; rd[r] += rd[r + o]; }
        __syncthreads();
    }
    float np = sqrtf(rp[0]), ns = sqrtf(rs[0]), dot = rd[0];
    __syncthreads();
    float sim = dot / (fmaxf(np, 1e-6f) * fmaxf(ns, 1e-6f));
    sim = fminf(1.0f, fmaxf(-1.0f, sim));
    float gate = 0.1f + 0.8f * 0.5f * (sim + 1.0f);
    float evidence = fminf(1.0f, (float)(*ept) / 256.0f);
    gate *= evidence;
    float upd = (1.0f - gate) * prev + gate * snew;
    rp[r] = upd * upd; __syncthreads();
    for (int o = 32; o > 0; o >>= 1) {
        if (r < o) rp[r] += rp[r + o];
        __syncthreads();
    }
    float n = sqrtf(rp[0]);
    float scale = fminf(1.0f, 10.0f / fmaxf(n, 1e-6f));
    out[b * Rsz + r] = upd * scale;
}

// ---------------------------------------------------------------------------
extern "C" void kernel_launch(void* const* d_in, const int* in_sizes, int n_in,
                              void* d_out, int out_size, void* d_ws, size_t ws_size,
                              hipStream_t stream) {
    const float* state = (const float*)d_in[0];
    const float* K     = (const float*)d_in[1];
    const float* V     = (const float*)d_in[2];
    const float* Wq    = (const float*)d_in[3];
    const float* Wk    = (const float*)d_in[4];
    const float* Wv    = (const float*)d_in[5];
    const float* gq    = (const float*)d_in[6];
    const float* lq    = (const float*)d_in[7];
    const int*   ept   = (const int*)d_in[8];
    float* out = (float*)d_out;

    // workspace layout (floats); total ~17.1 MB
    float* ws      = (float*)d_ws;
    float* wkq     = ws;              // 4096
    float* partial = ws + 4096;       // 1024
    float* layerw  = ws + 5120;       // 64
    float* qmat    = ws + 5184;       // 1024
    float* ctx     = ws + 6208;       // 2048
    float* kmat    = ws + 8256;       // 32768
    float* vmat    = ws + 41024;      // 32768
    float* ekbuf   = ws + 73792;      // 2097152
    float* evbuf   = ws + 2170944;    // 2097152

    wkq_kernel<<<Hsz / 256, 256, 0, stream>>>(Wk, lq, wkq);
    score_partial_kernel<<<dim3(CHUNKS, Bsz * Lsz), 256, 0, stream>>>(K, wkq, partial);
    layer_softmax_kernel<<<1, Bsz * Lsz, 0, stream>>>(partial, layerw);
    weighted_sum_kernel<<<2048, 256, 0, stream>>>(K, V, layerw, ekbuf, evbuf);
    proj_wmma_kernel<<<(Bsz * Esz) / 16, 128, 0, stream>>>(ekbuf, Wk, kmat);
    proj_wmma_kernel<<<(Bsz * Esz) / 16, 128, 0, stream>>>(evbuf, Wv, vmat);
    proj_wmma_kernel<<<1, 128, 0, stream>>>(gq, Wq, qmat);
    attn_kernel<<<Bsz * Gsz, 256, 0, stream>>>(qmat, kmat, vmat, ctx);
    state_update_kernel<<<Bsz, Rsz, 0, stream>>>(ctx, state, ept, out);
}